// GraphPool_721554506558
// MI455X (gfx1250) — compile-verified
//
#include <hip/hip_runtime.h>
#include <hip/hip_bf16.h>
#include <stdint.h>

#define B 16
#define N 50000
#define D 128
#define KKEEP 25000
#define NBIN 65536
#define TILES_PER_BATCH (N / 16)   // 3125

typedef __attribute__((ext_vector_type(2))) float v2f;
typedef __attribute__((ext_vector_type(8))) float v8f;

// Monotone float -> u32 map (order-preserving for all finite floats).
__device__ __forceinline__ unsigned key_of(float s) {
  unsigned u = __float_as_uint(s);
  return (u & 0x80000000u) ? ~u : (u | 0x80000000u);
}

__global__ void k_zero(unsigned* __restrict__ p, int n) {
  int i = blockIdx.x * blockDim.x + threadIdx.x;
  if (i < n) p[i] = 0u;
}

// One wave (32 lanes) computes scores for 16 consecutive nodes of one batch
// using V_WMMA_F32_16X16X4_F32 accumulation over the 128-dim contraction.
__global__ void __launch_bounds__(32) k_scores(
    const float* __restrict__ h, const float* __restrict__ W,
    const float* __restrict__ bias,
    float* __restrict__ scores, unsigned* __restrict__ keys)
{
  __shared__ float tile[16 * 132];   // 16 rows x 128 cols, padded to 132 (conflict-free)
  __shared__ float wsh[128];

  const int lane    = threadIdx.x;
  const int t       = blockIdx.x;
  const int b       = t / TILES_PER_BATCH;
  const int rowbase = (t % TILES_PER_BATCH) * 16;
  const float* src  = h + ((size_t)b * N + rowbase) * D;

  // Coalesced load of the 16x128 tile into padded LDS.
  #pragma unroll
  for (int c = 0; c < 16; ++c) {
    float4 v = *(const float4*)(src + c * 128 + lane * 4);
    *(float4*)&tile[c * 132 + lane * 4] = v;
  }
  *(float4*)&wsh[lane * 4] = *(const float4*)(W + lane * 4);
  __syncthreads();

  // A-matrix 16x4 f32 layout (ISA 7.12.2): lanes 0-15 row M=lane hold K0,K1;
  // lanes 16-31 row M=lane-16 hold K2,K3. B: W chunk replicated across all 16
  // columns -> every column of D equals the 16 dot products.
  const int row  = lane & 15;
  const int koff = (lane >> 4) << 1;
  v8f acc = {};
  #pragma unroll
  for (int kb = 0; kb < 32; ++kb) {
    const int kbase = 4 * kb + koff;
    v2f a;  a.x  = tile[row * 132 + kbase]; a.y  = tile[row * 132 + kbase + 1];
    v2f bb; bb.x = wsh[kbase];              bb.y = wsh[kbase + 1];
    acc = __builtin_amdgcn_wmma_f32_16x16x4_f32(
        /*neg_a=*/false, a, /*neg_b=*/false, bb,
        /*c_mod=*/(short)0, acc, /*reuse_a=*/false, /*reuse_b=*/false);
  }

  // D layout: lane l<16 -> (M=vgpr, N=l), lane l>=16 -> (M=vgpr+8, N=l-16).
  // Columns are replicated, so lane 0 holds rows 0-7, lane 16 rows 8-15.
  if ((lane & 15) == 0) {
    const float bv = bias[0];
    const int mbase = (lane >> 4) * 8;
    #pragma unroll
    for (int i = 0; i < 8; ++i) {
      float x = acc[i] + bv;
      float s = 1.0f / (1.0f + __expf(-x));
      size_t o = (size_t)b * N + rowbase + mbase + i;
      scores[o] = s;
      keys[o]   = key_of(s);
    }
  }
}

__global__ void k_hist(const unsigned* __restrict__ keys, unsigned* __restrict__ hist) {
  int i = blockIdx.x * blockDim.x + threadIdx.x;
  if (i < B * N) {
    int b = i / N;
    unsigned bin = keys[i] >> 16;
    atomicAdd(&hist[(size_t)b * NBIN + bin], 1u);
  }
}

// Per-batch suffix sum over 65536 bins: cum[bin] = #elements in bins > bin.
__global__ void __launch_bounds__(1024) k_scan(
    const unsigned* __restrict__ hist, unsigned* __restrict__ cum)
{
  __shared__ unsigned lsum[1024];
  __shared__ unsigned after[1024];
  const int b = blockIdx.x, t = threadIdx.x;
  const unsigned* hb = hist + (size_t)b * NBIN;
  unsigned* cb = cum + (size_t)b * NBIN;

  unsigned s = 0;
  for (int i = 0; i < 64; ++i) s += hb[t * 64 + i];
  lsum[t] = s;
  __syncthreads();
  if (t == 0) {
    unsigned run = 0;
    for (int u = 1023; u >= 0; --u) { after[u] = run; run += lsum[u]; }
  }
  __syncthreads();
  unsigned run = after[t];
  for (int i = 63; i >= 0; --i) {
    int bin = t * 64 + i;
    cb[bin] = run;
    run += hb[bin];
  }
}

// Scatter node ids into per-bin lists (descending-bin layout at cum[bin]).
// Arrival order within a bin is nondeterministic; the SET per bin is not,
// and downstream ranking depends only on the set.
__global__ void k_scatter(const unsigned* __restrict__ keys,
                          const unsigned* __restrict__ cum,
                          unsigned* __restrict__ cursor,
                          unsigned* __restrict__ list)
{
  int i = blockIdx.x * blockDim.x + threadIdx.x;
  if (i < B * N) {
    int b = i / N;
    int n = i % N;
    unsigned bin = keys[i] >> 16;
    unsigned pos = cum[(size_t)b * NBIN + bin] +
                   atomicAdd(&cursor[(size_t)b * NBIN + bin], 1u);
    list[(size_t)b * N + pos] = (unsigned)n;
  }
}

// One wave per node: every lane redundantly computes the exact rank
// (wave-uniform, deterministic: value+index comparisons only), then the
// selected rows are copied with one 128b load/store per lane (512B/row).
__global__ void __launch_bounds__(256) k_gather(
    const float* __restrict__ h, const float* __restrict__ scores,
    const unsigned* __restrict__ keys, const unsigned* __restrict__ hist,
    const unsigned* __restrict__ cum, const unsigned* __restrict__ list,
    float* __restrict__ out)
{
  size_t gtid = (size_t)blockIdx.x * blockDim.x + threadIdx.x;
  int lane = threadIdx.x & 31;
  size_t e = gtid >> 5;
  if (e >= (size_t)B * N) return;
  int b = (int)(e / N);
  int n = (int)(e % N);

  unsigned myKey = keys[e];
  unsigned bin   = myKey >> 16;
  unsigned cA    = cum[(size_t)b * NBIN + bin];
  unsigned cnt   = hist[(size_t)b * NBIN + bin];
  unsigned r     = cA;
  const unsigned* lb = list + (size_t)b * N + cA;
  for (unsigned i = 0; i < cnt; ++i) {
    unsigned m = lb[i];
    if ((int)m == n) continue;
    unsigned km = keys[(size_t)b * N + m];
    if (km > myKey || (km == myKey && (int)m < n)) ++r;
  }

  if (r < KKEEP) {
    float s = scores[e];
    float4 v = *((const float4*)(h + e * (size_t)D) + lane);
    v.x *= s; v.y *= s; v.z *= s; v.w *= s;
    *((float4*)(out + ((size_t)b * KKEEP + r) * D) + lane) = v;
  }
}

extern "C" void kernel_launch(void* const* d_in, const int* in_sizes, int n_in,
                              void* d_out, int out_size, void* d_ws, size_t ws_size,
                              hipStream_t stream) {
  const float* h    = (const float*)d_in[0];
  const float* W    = (const float*)d_in[1];
  const float* bias = (const float*)d_in[2];
  float* out = (float*)d_out;

  uint8_t* ws = (uint8_t*)d_ws;
  const size_t BN = (size_t)B * N;
  unsigned* keys   = (unsigned*)ws;                       // BN u32
  float*    scores = (float*)(ws + BN * 4);               // BN f32
  unsigned* hist   = (unsigned*)(ws + BN * 8);            // B*NBIN u32
  unsigned* cum    = hist + (size_t)B * NBIN;             // B*NBIN u32
  unsigned* cursor = cum  + (size_t)B * NBIN;             // B*NBIN u32
  unsigned* list   = cursor + (size_t)B * NBIN;           // BN u32

  // Zero hist + cum + cursor (contiguous); cum is fully rewritten by k_scan.
  int zn = 3 * B * NBIN;
  k_zero<<<(zn + 255) / 256, 256, 0, stream>>>(hist, zn);

  k_scores<<<B * TILES_PER_BATCH, 32, 0, stream>>>(h, W, bias, scores, keys);
  k_hist<<<(B * N + 255) / 256, 256, 0, stream>>>(keys, hist);
  k_scan<<<B, 1024, 0, stream>>>(hist, cum);
  k_scatter<<<(B * N + 255) / 256, 256, 0, stream>>>(keys, cum, cursor, list);

  // B*N waves of 32 lanes = B*N*32 threads; 256 threads/block.
  size_t gthreads = (size_t)B * N * 32;
  k_gather<<<(unsigned)((gthreads + 255) / 256), 256, 0, stream>>>(
      h, scores, keys, hist, cum, list, out);
}